// BoxCrop_9311489098114
// MI455X (gfx1250) — compile-verified
//
#include <hip/hip_runtime.h>
#include <hip/hip_bf16.h>

// BoxCrop: fused crop -> aspect-preserving bilinear resize (long side=336)
// -> square pad (fill=127).  images: [B,3,768,768] f32, boxes: [B,4] i32 (XYWH)
// out: [B,3,336,336] f32.
//
// Memory-bound gather kernel (AI ~0.1 FLOP/byte): no matrix contraction, so
// WMMA/TDM are inapplicable.  Optimized for MI455X as: scalar (SGPR) per-image
// box state, wave32-coalesced output stores with non-temporal hint (keep the
// 192MB L2 for reused bilinear source lines), shared gather offsets across the
// 3 channels, and load-free fill in padding regions.

#define OSZ   336
#define FILLV 127.0f
#define IMW   768
#define IMH   768
#define NCH   3

__global__ __launch_bounds__(256) void BoxCrop_kernel(
    const float* __restrict__ images,
    const int*   __restrict__ boxes,
    float*       __restrict__ out)
{
    const int b = blockIdx.y;
    const int p = blockIdx.x * 256 + threadIdx.x;     // pixel index within image
    if (p >= OSZ * OSZ) return;
    const int oy = p / OSZ;
    const int ox = p - oy * OSZ;

    // Uniform (blockIdx-only) address -> compiler emits s_load_b128; all the
    // per-image parameter math below stays in SALU/SGPRs.
    const int4 bx = reinterpret_cast<const int4*>(boxes)[b];
    const int xb = bx.x, yb = bx.y, wb = bx.z, hb = bx.w;

    const float wf = (float)wb, hf = (float)hb;
    const float scale = (float)OSZ / fmaxf(wf, hf);
    // jnp.round == round-half-even == rintf (default RNE mode)
    const int new_w = (int)rintf(wf * scale);
    const int new_h = (int)rintf(hf * scale);
    const int pad_top  = (hb <  wb) ? ((OSZ - new_h) >> 1) : 0;
    const int pad_left = (hb >= wb) ? ((OSZ - new_w) >> 1) : 0;

    const int iy = oy - pad_top;
    const int ix = ox - pad_left;

    float* o0 = out + ((size_t)b * NCH) * (size_t)(OSZ * OSZ)
                    + (size_t)oy * OSZ + ox;

    const bool valid = (iy >= 0) && (iy < new_h) && (ix >= 0) && (ix < new_w);
    if (!valid) {
        // Padding: warp-uniform regions; skip all gathers, stream NT fill.
        __builtin_nontemporal_store(FILLV, o0);
        __builtin_nontemporal_store(FILLV, o0 + OSZ * OSZ);
        __builtin_nontemporal_store(FILLV, o0 + 2 * OSZ * OSZ);
        return;
    }

    // Match reference evaluation order: ((i + 0.5) * len) / new_len
    const float src_y = (float)yb + ((float)iy + 0.5f) * hf / (float)new_h - 0.5f;
    const float src_x = (float)xb + ((float)ix + 0.5f) * wf / (float)new_w - 0.5f;
    const float y0f = floorf(src_y);
    const float x0f = floorf(src_x);
    const float wy  = src_y - y0f;
    const float wx  = src_x - x0f;

    const int ylo = yb, yhi = yb + hb - 1;
    const int xlo = xb, xhi = xb + wb - 1;
    const int y0 = min(max((int)y0f,     ylo), yhi);
    const int y1 = min(max((int)y0f + 1, ylo), yhi);
    const int x0 = min(max((int)x0f,     xlo), xhi);
    const int x1 = min(max((int)x0f + 1, xlo), xhi);

    // Channel-independent gather offsets (computed once, reused x3).
    const int o00 = y0 * IMW + x0;
    const int o01 = y0 * IMW + x1;
    const int o10 = y1 * IMW + x0;
    const int o11 = y1 * IMW + x1;

    const float wx1 = 1.0f - wx;
    const float wy1 = 1.0f - wy;

    const float* img = images + (size_t)b * (NCH * IMH * IMW);

#pragma unroll
    for (int c = 0; c < NCH; ++c) {
        const float* ch = img + (size_t)c * (IMH * IMW);
        const float v00 = ch[o00];
        const float v01 = ch[o01];
        const float v10 = ch[o10];
        const float v11 = ch[o11];
        const float top = v00 * wx1 + v01 * wx;
        const float bot = v10 * wx1 + v11 * wx;
        const float val = top * wy1 + bot * wy;
        // Write-once output: non-temporal so it doesn't evict reused source
        // lines from L2.
        __builtin_nontemporal_store(val, o0 + c * (OSZ * OSZ));
    }
}

extern "C" void kernel_launch(void* const* d_in, const int* in_sizes, int n_in,
                              void* d_out, int out_size, void* d_ws, size_t ws_size,
                              hipStream_t stream)
{
    const float* images = (const float*)d_in[0];
    const int*   boxes  = (const int*)d_in[1];
    float*       out    = (float*)d_out;

    const int B = in_sizes[1] / 4;                 // boxes is [B,4]
    const int pix_blocks = (OSZ * OSZ + 255) / 256; // 441 (exact)

    dim3 grid(pix_blocks, B);
    dim3 block(256);
    BoxCrop_kernel<<<grid, block, 0, stream>>>(images, boxes, out);
}